// HierProtoPNet_50568944943235
// MI455X (gfx1250) — compile-verified
//
#include <hip/hip_runtime.h>

typedef __attribute__((ext_vector_type(2))) float v2f;
typedef __attribute__((ext_vector_type(8))) float v8f;
typedef __attribute__((ext_vector_type(4))) unsigned int v4u;
typedef __attribute__((ext_vector_type(4))) int v4i;
typedef __attribute__((ext_vector_type(8))) int v8i;

#define PCNT 2000
#define KCNT 200
#define BATCH 32
#define CDIM 128

// Each 16-column half-tile lives in its own LDS region: 128 channel rows,
// row stride 24 dwords (16 data + 8 TDM pad) -> conflict-free 2addr reads,
// and regions are 12 KB apart so the DS load-merger can only pair the
// K-operand dwords (row, row+1 @ same column) into consecutive VGPRs.
#define ROWSTRIDE 24
#define HALFDW (CDIM * ROWSTRIDE)  // 3072 dwords = 12 KB per half-tile region

// exp(-d/256) = exp2(-d * log2(e) / 256)
__device__ __forceinline__ float sim_exp(float d) {
  return __builtin_amdgcn_exp2f(d * (-1.44269504088896340736f / 256.0f));
}

__global__ void p2_kernel(const float* __restrict__ proto, float* __restrict__ p2) {
  int p = blockIdx.x * blockDim.x + threadIdx.x;
  if (p >= PCNT) return;
  const float4* row = (const float4*)(proto + (size_t)p * CDIM);
  float s = 0.f;
#pragma unroll 8
  for (int i = 0; i < CDIM / 4; ++i) {
    float4 v = row[i];
    s += v.x * v.x + v.y * v.y + v.z * v.z + v.w * v.w;
  }
  p2[p] = s;
}

template <int HW>
__global__ void x2_kernel(const float* __restrict__ x, float* __restrict__ x2) {
  int i = blockIdx.x * blockDim.x + threadIdx.x;
  if (i >= BATCH * HW) return;
  int b = i / HW, hw = i - b * HW;
  const float* p = x + (size_t)b * CDIM * HW + hw;
  float s = 0.f;
#pragma unroll 8
  for (int c = 0; c < CDIM; ++c) {
    float v = p[(size_t)c * HW];
    s += v * v;
  }
  x2[i] = s;
}

// TDM: stage a [CDIM x 16] x half-tile (4B elements) into LDS at `ldsaddr`.
// dim0_rem clamps the ragged column tail (TDM OOB reads zero-fill).
// pad_enable inserts 8 dwords after every 16 -> LDS row stride 24 dwords.
__device__ __forceinline__ void tdm_load_half(const float* gsrc, unsigned ldsaddr,
                                              unsigned dim0_rem, unsigned stride) {
  unsigned long long ga = (unsigned long long)(const void*)gsrc;
  v4u g0;
  g0[0] = 1u;                                   // count=1 (valid), user mode
  g0[1] = ldsaddr;                              // lds_addr (bytes)
  g0[2] = (unsigned)(ga & 0xFFFFFFFFull);       // global_addr[31:0]
  g0[3] = (unsigned)((ga >> 32) & 0x1FFFFFFull) // global_addr[56:32]
          | (2u << 30);                         // type = 2 ("image")
  v8i g1;
  g1[0] = (2 << 16)      // data_size = 4 bytes
          | (1 << 20)    // pad_enable
          | (3 << 22)    // pad_interval: every 16 dwords
          | (7 << 25);   // pad_amount: 8 dwords
  g1[1] = (int)((dim0_rem & 0xFFFFu) << 16);       // tensor_dim0[15:0]
  g1[2] = (int)((dim0_rem >> 16) | (128u << 16));  // dim0[31:16] | tensor_dim1=128
  g1[3] = (int)(16u << 16);                        // tile_dim0 = 16
  g1[4] = 128;                                     // tile_dim1 = 128
  g1[5] = (int)stride;                             // tensor_dim0_stride = HW
  g1[6] = 0;
  g1[7] = 0;
  v4i z4 = {0, 0, 0, 0};
#if __clang_major__ >= 23
  v8i z8 = {0, 0, 0, 0, 0, 0, 0, 0};
  __builtin_amdgcn_tensor_load_to_lds(g0, g1, z4, z4, z8, 0);
#else
  __builtin_amdgcn_tensor_load_to_lds(g0, g1, z4, z4, 0);
#endif
}

// Issue both half-tiles of the 32-wide column tile at hw0 into buffer `buf`.
template <int HW>
__device__ __forceinline__ void tdm_load_tile(const float* xb, unsigned lds0,
                                              int buf, int hw0) {
  unsigned base = lds0 + (unsigned)(buf * 2 * HALFDW * 4);
  tdm_load_half(xb + hw0, base, (unsigned)(HW - hw0), (unsigned)HW);
  int rem1 = HW - hw0 - 16;
  tdm_load_half(xb + hw0 + 16, base + HALFDW * 4,
                (unsigned)(rem1 > 0 ? rem1 : 0), (unsigned)HW);
}

// Block = 5 waves = 80 prototype rows for one image b. Proto A-fragments are
// register-cached once (32 x v2f = 64 VGPRs); the block loops over all 32-wide
// column tiles of the image with double-buffered async TDM staging, so the
// steady-state inner loop is pure LDS + v_wmma_f32_16x16x4_f32 (two
// independent accumulator chains). Each (b,p) min is owned by one wave ->
// register min + one plain store at the end (no atomics, no init pass).
template <int HW>
__global__ __launch_bounds__(160) void proto_dist_kernel(
    const float* __restrict__ x, const float* __restrict__ proto,
    const float* __restrict__ x2, const float* __restrict__ p2,
    float* __restrict__ sim, float* __restrict__ mind) {
  constexpr int R = (HW + 31) / 32;  // column tiles per image
  __shared__ float xs[4 * HALFDW];   // 2 buffers x 2 half-tile regions
  const int b = blockIdx.y;
  const int t = threadIdx.x;
  const unsigned lds0 = (unsigned)(unsigned long long)(void*)xs;
  const float* xb = x + (size_t)b * CDIM * HW;

  const int wave = t >> 5;
  const int lane = t & 31;
  const int hi = lane >> 4;  // lane half
  const int lo = lane & 15;
  const int pbase = (blockIdx.x * 5 + wave) * 16;  // 25*5*16 == 2000 exactly

  // Preload A fragments (16x4 f32 layout): lane holds
  // proto[pbase+lo][4j + 2*hi + {0,1}] for all 32 K-steps.
  const float* arow = proto + (size_t)(pbase + lo) * CDIM + 2 * hi;
  v2f a[32];
#pragma unroll
  for (int j = 0; j < 32; ++j) a[j] = *(const v2f*)(arow + 4 * j);

  const int prow0 = pbase + 8 * hi;
  float4 p2a = *(const float4*)(p2 + prow0);
  float4 p2b = *(const float4*)(p2 + prow0 + 4);
  float p2v[8] = {p2a.x, p2a.y, p2a.z, p2a.w, p2b.x, p2b.y, p2b.z, p2b.w};
  float dmr[8];
#pragma unroll
  for (int r = 0; r < 8; ++r) dmr[r] = __builtin_inff();

  if (t < 32)  // wave 0: prologue TDM for tile 0 -> buffer 0
    tdm_load_tile<HW>(xb, lds0, 0, 0);

  for (int it = 0; it < R; ++it) {
    if (t < 32) __builtin_amdgcn_s_wait_tensorcnt(0);  // tile `it` landed
    __syncthreads();  // all waves: tile ready; prev compute finished
    asm volatile("" ::: "memory");  // TDM wrote LDS; keep xs reads honest
    if (t < 32 && it + 1 < R)  // async-stage next tile into other buffer
      tdm_load_tile<HW>(xb, lds0, (it + 1) & 1, (it + 1) * 32);

    const int reg0 = (it & 1) * 2 * HALFDW;
    v8f acc0 = {0.f, 0.f, 0.f, 0.f, 0.f, 0.f, 0.f, 0.f};
    v8f acc1 = acc0;
#pragma unroll
    for (int j = 0; j < 32; ++j) {
      const int r0 = reg0 + (4 * j + 2 * hi) * ROWSTRIDE + lo;
      v2f b0 = {xs[r0], xs[r0 + ROWSTRIDE]};                      // N-tile 0
      v2f b1 = {xs[r0 + HALFDW], xs[r0 + HALFDW + ROWSTRIDE]};    // N-tile 1
      acc0 = __builtin_amdgcn_wmma_f32_16x16x4_f32(false, a[j], false, b0,
                                                   (short)0, acc0, false, false);
      acc1 = __builtin_amdgcn_wmma_f32_16x16x4_f32(false, a[j], false, b1,
                                                   (short)0, acc1, false, false);
    }

    // Epilogue: C/D layout -> VGPR r, lane: M = r + 8*hi, N = lo (per tile)
    const int col0 = it * 32 + lo;
    const int col1 = col0 + 16;
    const bool v0 = col0 < HW;
    const bool v1 = col1 < HW;
    const float xx0 = v0 ? x2[b * HW + col0] : 0.0f;
    const float xx1 = v1 ? x2[b * HW + col1] : 0.0f;
#pragma unroll
    for (int r = 0; r < 8; ++r) {
      float d0 = fmaxf(xx0 - 2.0f * acc0[r] + p2v[r], 0.0f);
      float d1 = fmaxf(xx1 - 2.0f * acc1[r] + p2v[r], 0.0f);
      const size_t rowoff = ((size_t)b * PCNT + prow0 + r) * HW;
      if (v0) __builtin_nontemporal_store(sim_exp(d0), sim + rowoff + col0);
      if (v1) __builtin_nontemporal_store(sim_exp(d1), sim + rowoff + col1);
      float dm01 = fminf(v0 ? d0 : __builtin_inff(), v1 ? d1 : __builtin_inff());
      dmr[r] = fminf(dmr[r], dm01);
    }
  }

  // Publish min: reduce within each 16-lane half; this wave owns these rows.
#pragma unroll
  for (int r = 0; r < 8; ++r) {
    float dm = dmr[r];
#pragma unroll
    for (int m = 1; m <= 8; m <<= 1)
      dm = fminf(dm, __shfl_xor(dm, m, 32));
    if (lo == 0) mind[(size_t)b * PCNT + prow0 + r] = dm;
  }
}

__global__ __launch_bounds__(256) void logits_kernel(
    const float* __restrict__ mind, const float* __restrict__ w,
    float* __restrict__ logits) {
  __shared__ float act[PCNT];
  const int b = blockIdx.x;
  for (int p = threadIdx.x; p < PCNT; p += 256)
    act[p] = sim_exp(mind[(size_t)b * PCNT + p]);
  __syncthreads();
  const int k = threadIdx.x;
  if (k < KCNT) {
    const float* wr = w + (size_t)k * PCNT;
    float s = 0.f;
#pragma unroll 4
    for (int p = 0; p < PCNT; ++p) s += act[p] * wr[p];
    logits[(size_t)b * KCNT + k] = s;
  }
}

extern "C" void kernel_launch(void* const* d_in, const int* in_sizes, int n_in,
                              void* d_out, int out_size, void* d_ws,
                              size_t ws_size, hipStream_t stream) {
  const float* x_h = (const float*)d_in[0];
  const float* x_m = (const float*)d_in[1];
  const float* x_l = (const float*)d_in[2];
  const float* pr_h = (const float*)d_in[3];
  const float* pr_m = (const float*)d_in[4];
  const float* pr_l = (const float*)d_in[5];
  const float* w_h = (const float*)d_in[6];
  const float* w_m = (const float*)d_in[7];
  const float* w_l = (const float*)d_in[8];
  float* out = (float*)d_out;

  // d_out flat layout (floats), in reference return order
  constexpr size_t LOG_H = 0;
  constexpr size_t LOG_M = LOG_H + (size_t)BATCH * KCNT;
  constexpr size_t LOG_L = LOG_M + (size_t)BATCH * KCNT;
  constexpr size_t MIN_H = LOG_L + (size_t)BATCH * KCNT;
  constexpr size_t MIN_M = MIN_H + (size_t)BATCH * PCNT;
  constexpr size_t MIN_L = MIN_M + (size_t)BATCH * PCNT;
  constexpr size_t SIM_H = MIN_L + (size_t)BATCH * PCNT;
  constexpr size_t SIM_M = SIM_H + (size_t)BATCH * PCNT * 49;
  constexpr size_t SIM_L = SIM_M + (size_t)BATCH * PCNT * 196;

  // workspace: p2 (3*2000) + x2 (32*(49+196+784)) = ~156 KB
  float* ws = (float*)d_ws;
  float* p2h = ws;
  float* p2m = ws + PCNT;
  float* p2l = ws + 2 * PCNT;
  float* x2h = ws + 3 * PCNT;
  float* x2m = x2h + BATCH * 49;
  float* x2l = x2m + BATCH * 196;

  p2_kernel<<<(PCNT + 255) / 256, 256, 0, stream>>>(pr_h, p2h);
  p2_kernel<<<(PCNT + 255) / 256, 256, 0, stream>>>(pr_m, p2m);
  p2_kernel<<<(PCNT + 255) / 256, 256, 0, stream>>>(pr_l, p2l);

  x2_kernel<49><<<(BATCH * 49 + 255) / 256, 256, 0, stream>>>(x_h, x2h);
  x2_kernel<196><<<(BATCH * 196 + 255) / 256, 256, 0, stream>>>(x_m, x2m);
  x2_kernel<784><<<(BATCH * 784 + 255) / 256, 256, 0, stream>>>(x_l, x2l);

  proto_dist_kernel<49><<<dim3(25, BATCH), 160, 0, stream>>>(
      x_h, pr_h, x2h, p2h, out + SIM_H, out + MIN_H);
  proto_dist_kernel<196><<<dim3(25, BATCH), 160, 0, stream>>>(
      x_m, pr_m, x2m, p2m, out + SIM_M, out + MIN_M);
  proto_dist_kernel<784><<<dim3(25, BATCH), 160, 0, stream>>>(
      x_l, pr_l, x2l, p2l, out + SIM_L, out + MIN_L);

  logits_kernel<<<BATCH, 256, 0, stream>>>(out + MIN_H, w_h, out + LOG_H);
  logits_kernel<<<BATCH, 256, 0, stream>>>(out + MIN_M, w_m, out + LOG_M);
  logits_kernel<<<BATCH, 256, 0, stream>>>(out + MIN_L, w_l, out + LOG_L);
}